// ATTConv_10290741641559
// MI455X (gfx1250) — compile-verified
//
#include <hip/hip_runtime.h>
#include <cstdint>

#define TT 3
#define DD 128
#define NEG_SLOPE 0.01f

typedef __attribute__((ext_vector_type(4))) float v4f;

__global__ __launch_bounds__(256) void attconv_kernel(
    const float* __restrict__ hc,   // (T,B,D)
    const float* __restrict__ hn,   // (T,T,B,D)
    const float* __restrict__ aw,   // (T,2D)
    const float* __restrict__ ab,   // (T,)
    float* __restrict__ out,        // (T,B,D)
    int Bn)
{
    __shared__ __align__(16) float s_w[TT * 2 * DD]; // 768 floats = 3 KB
    __shared__ float s_b[TT];

    const int tid  = threadIdx.x;
    const int lane = tid & 31;

    // ---- Stage att_w into LDS with CDNA5 async global->LDS DMA ----
    // 768 floats = 192 x 16B transfers; threads 0..191 = waves 0..5 fully
    // active (EXEC all-ones for the issuing waves).
    if (tid < 192) {
        const float* g = aw + tid * 4;
        unsigned ldsOff = (unsigned)(uintptr_t)(&s_w[tid * 4]); // low 32 bits = LDS byte offset
        asm volatile("global_load_async_to_lds_b128 %0, %1, off"
                     :: "v"(ldsOff), "v"(g) : "memory");
    }
    if (tid < TT) s_b[tid] = ab[tid];
    asm volatile("s_wait_asynccnt 0" ::: "memory");
    __syncthreads();

    const int wavesPerBlk = blockDim.x >> 5;
    const int wid    = blockIdx.x * wavesPerBlk + (tid >> 5);
    const int nwaves = gridDim.x * wavesPerBlk;
    const int rows   = TT * Bn;
    const int d0     = lane * 4;
    const size_t planeND = (size_t)Bn * DD;   // stride between neighbor planes

    for (int row = wid; row < rows; row += nwaves) {
        const int t = row / Bn;
        const int b = row - t * Bn;

        // Prefetch next row this wave will process (global_prefetch_b8).
        const int nrow = row + nwaves;
        if (nrow < rows) {
            const int nt = nrow / Bn;
            const int nb = nrow - nt * Bn;
            __builtin_prefetch(hc + (size_t)nrow * DD + d0, 0, 0);
            const size_t pb = ((size_t)(nt * TT) * Bn + nb) * DD + d0;
            __builtin_prefetch(hn + pb, 0, 0);
            __builtin_prefetch(hn + pb + planeND, 0, 0);
            __builtin_prefetch(hn + pb + 2 * planeND, 0, 0);
        }

        // Coalesced b128 loads: one full D=128 row per wave per stream.
        const size_t cbase = (size_t)row * DD + d0;
        const v4f c4 = *(const v4f*)(hc + cbase);
        const size_t nbase = ((size_t)(t * TT) * Bn + b) * DD + d0;
        const v4f n0 = *(const v4f*)(hn + nbase);
        const v4f n1 = *(const v4f*)(hn + nbase + planeND);
        const v4f n2 = *(const v4f*)(hn + nbase + 2 * planeND);

        // Weights from LDS (ds_load_b128).
        const v4f wh = *(const v4f*)(&s_w[t * (2 * DD) + d0]);
        const v4f we = *(const v4f*)(&s_w[t * (2 * DD) + DD + d0]);
        const float bt = s_b[t];

        // Per-lane partial dot products (4 elements each).
        float ph = c4.x * wh.x + c4.y * wh.y + c4.z * wh.z + c4.w * wh.w;
        float p0 = n0.x * we.x + n0.y * we.y + n0.z * we.z + n0.w * we.w;
        float p1 = n1.x * we.x + n1.y * we.y + n1.z * we.z + n1.w * we.w;
        float p2 = n2.x * we.x + n2.y * we.y + n2.z * we.z + n2.w * we.w;
        float p3 = c4.x * we.x + c4.y * we.y + c4.z * we.z + c4.w * we.w;

        // Wave32 butterfly reduction: all lanes end with the full sums.
        #pragma unroll
        for (int off = 16; off >= 1; off >>= 1) {
            ph += __shfl_xor(ph, off, 32);
            p0 += __shfl_xor(p0, off, 32);
            p1 += __shfl_xor(p1, off, 32);
            p2 += __shfl_xor(p2, off, 32);
            p3 += __shfl_xor(p3, off, 32);
        }

        // scores = score_e + score_h + bias, then LeakyReLU.
        float s0 = p0 + ph + bt;
        float s1 = p1 + ph + bt;
        float s2 = p2 + ph + bt;
        float s3 = p3 + ph + bt;
        s0 = (s0 >= 0.f) ? s0 : NEG_SLOPE * s0;
        s1 = (s1 >= 0.f) ? s1 : NEG_SLOPE * s1;
        s2 = (s2 >= 0.f) ? s2 : NEG_SLOPE * s2;
        s3 = (s3 >= 0.f) ? s3 : NEG_SLOPE * s3;

        // Softmax over the 4 neighbors (n=3 is the center row).
        const float m  = fmaxf(fmaxf(s0, s1), fmaxf(s2, s3));
        const float e0 = __expf(s0 - m);
        const float e1 = __expf(s1 - m);
        const float e2 = __expf(s2 - m);
        const float e3 = __expf(s3 - m);
        const float inv = 1.0f / (e0 + e1 + e2 + e3);
        const float a0 = e0 * inv, a1 = e1 * inv, a2 = e2 * inv, a3 = e3 * inv;

        v4f o;
        o.x = a0 * n0.x + a1 * n1.x + a2 * n2.x + a3 * c4.x;
        o.y = a0 * n0.y + a1 * n1.y + a2 * n2.y + a3 * c4.y;
        o.z = a0 * n0.z + a1 * n1.z + a2 * n2.z + a3 * c4.z;
        o.w = a0 * n0.w + a1 * n1.w + a2 * n2.w + a3 * c4.w;

        // Write-once output: non-temporal store, keep L2 for the input streams.
        __builtin_nontemporal_store(o, (v4f*)(out + cbase));
    }
}

extern "C" void kernel_launch(void* const* d_in, const int* in_sizes, int n_in,
                              void* d_out, int out_size, void* d_ws, size_t ws_size,
                              hipStream_t stream) {
    const float* hc = (const float*)d_in[0];   // h_center (T,B,D)
    const float* hn = (const float*)d_in[1];   // h_neigh  (T,T,B,D)
    const float* aw = (const float*)d_in[2];   // att_w    (T,2D)
    const float* ab = (const float*)d_in[3];   // att_b    (T,)
    float* out = (float*)d_out;

    const int Bn   = in_sizes[0] / (TT * DD);  // B = 100000
    const int rows = TT * Bn;                  // one wave32 per row
    const int wavesPerBlock = 256 / 32;
    const int rowsPerWave   = 2;               // give the prefetch work to overlap
    int blocks = (rows + wavesPerBlock * rowsPerWave - 1) / (wavesPerBlock * rowsPerWave);
    if (blocks < 1) blocks = 1;

    attconv_kernel<<<blocks, 256, 0, stream>>>(hc, hn, aw, ab, out, Bn);
}